// GNN_Layer_8967891714671
// MI455X (gfx1250) — compile-verified
//
#include <hip/hip_runtime.h>

// GNN layer: out = ReLU( X @ W_self^T + segment_sum((X @ W_neigh^T)[src], dst) + bias )
// N_NODES=50000, N_EDGES=800000, IN=OUT=128, all fp32.
//
// MI455X plan:
//  - Kernel 1 (WMMA f32 16x16x4): per 16-node tile, 8 waves each own a 16-wide
//    out-dim slice; each wave runs BOTH GEMMs (W_self, W_neigh) sharing the A
//    fragment. Writes h_self+bias -> d_out, X@W_neigh^T -> d_ws.
//  - Kernel 2: one wave per edge, 4 floats/lane, global_atomic_add_f32 into
//    d_out. Both 25.6MB tables are L2-resident (192MB L2) -> L2-rate scatter.
//  - Kernel 3: ReLU in place.

typedef __attribute__((ext_vector_type(2))) float v2f;
typedef __attribute__((ext_vector_type(8))) float v8f;

#define IN_DIM 128
#define OUT_DIM 128

// grid.x = ceil(n_nodes/16), block = 256 (8 waves); wave w covers out dims [16w,16w+16)
__global__ __launch_bounds__(256)
void gnn_gemm_kernel(const float* __restrict__ X,
                     const float* __restrict__ Wself,
                     const float* __restrict__ Wneigh,
                     const float* __restrict__ bias,
                     float* __restrict__ out,   // h_self + bias
                     float* __restrict__ xw,    // X @ W_neigh^T
                     int n_nodes)
{
    const int wave = threadIdx.x >> 5;   // 0..7 -> out-dim tile
    const int lane = threadIdx.x & 31;
    const int half = lane >> 4;          // 0|1
    const int l16  = lane & 15;
    const int mrow = blockIdx.x * 16;    // node-tile base
    const int ncol = wave * 16;          // out-dim base

    // A fragment rows: clamp so every lane reads valid memory (EXEC must stay
    // all-ones for WMMA); stores are guarded instead.
    int arow = mrow + l16;
    if (arow >= n_nodes) arow = n_nodes - 1;
    const float* xrow  = X      + (size_t)arow * IN_DIM;
    // B[k][n] = W[n][k]: lane's n = ncol+l16, contiguous K pairs in W row.
    const float* wsrow = Wself  + (size_t)(ncol + l16) * IN_DIM;
    const float* wnrow = Wneigh + (size_t)(ncol + l16) * IN_DIM;

    v8f acc_s = {};
    v8f acc_n = {};
    // K loop: 128 / 4 = 32 WMMA steps per matrix.
    for (int k0 = 0; k0 < IN_DIM; k0 += 4) {
        const int kk = k0 + 2 * half;    // lanes 0-15: K={k0,k0+1}; 16-31: K={k0+2,k0+3}
        v2f a  = *(const v2f*)(xrow  + kk);
        v2f bs = *(const v2f*)(wsrow + kk);
        v2f bn = *(const v2f*)(wnrow + kk);
        acc_s = __builtin_amdgcn_wmma_f32_16x16x4_f32(
                    false, a, false, bs, (short)0, acc_s, false, false);
        acc_n = __builtin_amdgcn_wmma_f32_16x16x4_f32(
                    false, a, false, bn, (short)0, acc_n, false, false);
    }

    // C/D layout: VGPR r -> M = r + 8*half, N = l16
    const float bval = bias[ncol + l16];
    #pragma unroll
    for (int r = 0; r < 8; ++r) {
        const int m = mrow + r + 8 * half;
        if (m < n_nodes) {
            const size_t o = (size_t)m * OUT_DIM + ncol + l16;
            out[o] = acc_s[r] + bval;
            xw[o]  = acc_n[r];
        }
    }
}

// One wave per edge: gather 128 floats from xw[src], atomic-add into out[dst].
__global__ __launch_bounds__(256)
void gnn_scatter_kernel(const float* __restrict__ xw,
                        const int* __restrict__ src,
                        const int* __restrict__ dst,
                        float* __restrict__ out,
                        int n_edges)
{
    const int edge = blockIdx.x * (blockDim.x >> 5) + (threadIdx.x >> 5);
    const int lane = threadIdx.x & 31;
    if (edge >= n_edges) return;

    const int s = src[edge];   // uniform across wave -> scalar load
    const int d = dst[edge];

    const float4 v = *((const float4*)(xw + (size_t)s * OUT_DIM) + lane);
    float* dp = out + (size_t)d * OUT_DIM + lane * 4;
#if defined(__HIP_DEVICE_COMPILE__)
    unsafeAtomicAdd(dp + 0, v.x);   // -> global_atomic_add_f32 (no CAS loop)
    unsafeAtomicAdd(dp + 1, v.y);
    unsafeAtomicAdd(dp + 2, v.z);
    unsafeAtomicAdd(dp + 3, v.w);
#endif
}

__global__ __launch_bounds__(256)
void gnn_relu_kernel(float* __restrict__ out, int n4)
{
    const int i = blockIdx.x * blockDim.x + threadIdx.x;
    if (i < n4) {
        float4* p = (float4*)out + i;
        float4 v = *p;
        v.x = fmaxf(v.x, 0.0f);
        v.y = fmaxf(v.y, 0.0f);
        v.z = fmaxf(v.z, 0.0f);
        v.w = fmaxf(v.w, 0.0f);
        *p = v;
    }
}

extern "C" void kernel_launch(void* const* d_in, const int* in_sizes, int n_in,
                              void* d_out, int out_size, void* d_ws, size_t ws_size,
                              hipStream_t stream)
{
    // setup_inputs order: X, src, dst, W_self, W_neigh, bias
    const float* X      = (const float*)d_in[0];
    const int*   src    = (const int*)  d_in[1];
    const int*   dst    = (const int*)  d_in[2];
    const float* Wself  = (const float*)d_in[3];
    const float* Wneigh = (const float*)d_in[4];
    const float* bias   = (const float*)d_in[5];
    float*       out    = (float*)d_out;
    float*       xw     = (float*)d_ws;      // 50000*128*4 = 25.6 MB scratch

    const int n_nodes = in_sizes[0] / IN_DIM;
    const int n_edges = in_sizes[1];

    // 1) Fused dual GEMM: d_out = X@Wself^T + bias ; d_ws = X@Wneigh^T
    {
        dim3 grid((n_nodes + 15) / 16);
        gnn_gemm_kernel<<<grid, 256, 0, stream>>>(X, Wself, Wneigh, bias, out, xw, n_nodes);
    }
    // 2) Edge scatter-add (8 edges per 256-thread block)
    {
        dim3 grid((n_edges + 7) / 8);
        gnn_scatter_kernel<<<grid, 256, 0, stream>>>(xw, src, dst, out, n_edges);
    }
    // 3) ReLU in place
    {
        const int n4 = (n_nodes * OUT_DIM) / 4;
        dim3 grid((n4 + 255) / 256);
        gnn_relu_kernel<<<grid, 256, 0, stream>>>(out, n4);
    }
}